// AtomGroupBridgeFiLM_4088808866415
// MI455X (gfx1250) — compile-verified
//
#include <hip/hip_runtime.h>
#include <hip/hip_bf16.h>
#include <math.h>

// ---------------------------------------------------------------------------
// AtomGroupBridgeFiLM on MI455X (gfx1250, wave32, WMMA).
//
// Roofline: ~90 GFLOP of dense GEMM after exploiting the zero-initialized
// Set2Set (step-1 LSTM state is uniform across rows, so step-2 reduces to
// r1 @ Wih[:,GD:2GD]^T + const). Mandatory HBM traffic ~2GB -> ~90us floor at
// 23.3 TB/s, so fp32 WMMA (V_WMMA_F32_16X16X4_F32) comfortably exceeds the
// required ~1 TFLOP/s while preserving the reference's fp32 semantics.
//
// Round-1 fix: NT templated (killed v_movrel dynamic-VGPR indexing).
// Round-2 fix: fragment loads go through explicit addrspace(1) pointers so
// they lower to global_load_b64 (LOADcnt only) instead of flat_load_b64
// (which ticks LOADcnt+DScnt and does the LDS-aperture check).
// ---------------------------------------------------------------------------

#define NA_  200000
#define GM_  50000
#define NINC 400000
#define EGN  400000
#define AD_  128
#define GD_  160
#define CD_  64
#define G80  80

typedef __attribute__((ext_vector_type(2))) float v2f;
typedef __attribute__((ext_vector_type(8))) float v8f;
typedef const __attribute__((address_space(1))) float* gptr;

// 8-byte global load of a float pair, guaranteed global address space.
__device__ __forceinline__ v2f ld2g(const float* p) {
    gptr g = (gptr)p;
    v2f r; r.x = g[0]; r.y = g[1]; return r;
}

__device__ __forceinline__ float sigf(float x) { return 1.0f / (1.0f + expf(-x)); }
// order-preserving float<->uint encoding for atomicMax-based segment max
__device__ __forceinline__ unsigned enc_f(float f) {
    unsigned b = __float_as_uint(f);
    return (b & 0x80000000u) ? ~b : (b | 0x80000000u);
}
__device__ __forceinline__ float dec_f(unsigned k) {
    return (k & 0x80000000u) ? __uint_as_float(k ^ 0x80000000u) : __uint_as_float(~k);
}

// ---------------------------------------------------------------------------
// Small kernel: step-1 LSTM (uniform across rows: q_star=h=c=0) for both
// set2sets, plus gconst[n] = bih+bhh + h1·(WihL_row + Whh_row).
// out layout per set: h1[160] | c1[160] | gconst[640]  (960 floats), a then g.
// ---------------------------------------------------------------------------
__global__ void k_small_init(const float* __restrict__ Wih_a, const float* __restrict__ Whh_a,
                             const float* __restrict__ bih_a, const float* __restrict__ bhh_a,
                             const float* __restrict__ Wih_g, const float* __restrict__ Whh_g,
                             const float* __restrict__ bih_g, const float* __restrict__ bhh_g,
                             float* __restrict__ out) {
    __shared__ float h1s[GD_];
    const int t = threadIdx.x;
    for (int which = 0; which < 2; ++which) {
        const float* Wih = which ? Wih_g : Wih_a;
        const float* Whh = which ? Whh_g : Whh_a;
        const float* bih = which ? bih_g : bih_a;
        const float* bhh = which ? bhh_g : bhh_a;
        float* h1 = out + which * 960;
        float* c1 = h1 + GD_;
        float* gc = h1 + 2 * GD_;
        if (t < GD_) {
            float gi = bih[t] + bhh[t];
            float gg = bih[2 * GD_ + t] + bhh[2 * GD_ + t];
            float go = bih[3 * GD_ + t] + bhh[3 * GD_ + t];
            float c = sigf(gi) * tanhf(gg);   // sig(f)*c0 term vanishes (c0=0)
            float h = sigf(go) * tanhf(c);
            c1[t] = c; h1[t] = h; h1s[t] = h;
        }
        __syncthreads();
        for (int n = t; n < 4 * GD_; n += blockDim.x) {
            float acc = bih[n] + bhh[n];
            const float* wi = Wih + (size_t)n * (2 * GD_);   // WihL = cols 0..159
            const float* wh = Whh + (size_t)n * GD_;
            for (int k = 0; k < GD_; ++k) acc += h1s[k] * (wi[k] + wh[k]);
            gc[n] = acc;
        }
        __syncthreads();
    }
}

// ---------------------------------------------------------------------------
// Generic fp32 WMMA GEMM: C[M,N] = act(A[M,K] @ W[N,K]^T + bias [+ addend]).
// Optional split A (k < ksplit from A, else A2) for q_star=[h2|r2] inputs.
// 4 waves/block; wave = 16 rows x NT*16 cols; K stepped by 4 (all K%4==0).
// NT is a template parameter: accumulators statically indexed, W pointers in
// registers, all hot-loop addressing via pointer increments.
// ---------------------------------------------------------------------------
template <int NT>
__global__ __launch_bounds__(128)
void k_gemm(const float* __restrict__ A, int lda,
            const float* __restrict__ A2, int lda2, int ksplit,
            const float* __restrict__ Wt, int ldw,
            const float* __restrict__ bias,
            const float* __restrict__ addend, int ldadd,
            float* __restrict__ C, int ldc,
            int M, int N, int K, int act) {
    const int wave = threadIdx.x >> 5;
    const int lane = threadIdx.x & 31;
    const int m0 = blockIdx.x * 64 + wave * 16;
    if (m0 >= M) return;                       // whole-wave uniform exit
    const int n0 = blockIdx.y * (NT * 16);
    const int mrow = lane & 15;
    const int khalf = (lane >> 4) << 1;        // f32 16x16x4 A/B frag layout

    v8f acc[NT];
#pragma unroll
    for (int t = 0; t < NT; ++t)
#pragma unroll
        for (int v = 0; v < 8; ++v) acc[t][v] = 0.0f;

    const float* wp[NT];
#pragma unroll
    for (int t = 0; t < NT; ++t)
        wp[t] = Wt + (size_t)(n0 + t * 16 + mrow) * ldw + khalf;

    const int k1 = (A2 != nullptr) ? ksplit : K;
    const float* ap = A + (size_t)(m0 + mrow) * lda + khalf;
    for (int k = 0; k < k1; k += 4) {
        v2f af = ld2g(ap); ap += 4;
#pragma unroll
        for (int t = 0; t < NT; ++t) {
            v2f bf = ld2g(wp[t]); wp[t] += 4;
            acc[t] = __builtin_amdgcn_wmma_f32_16x16x4_f32(
                false, af, false, bf, (short)0, acc[t], false, false);
        }
    }
    if (A2 != nullptr) {
        ap = A2 + (size_t)(m0 + mrow) * lda2 + khalf;
        for (int k = k1; k < K; k += 4) {
            v2f af = ld2g(ap); ap += 4;
#pragma unroll
            for (int t = 0; t < NT; ++t) {
                v2f bf = ld2g(wp[t]); wp[t] += 4;
                acc[t] = __builtin_amdgcn_wmma_f32_16x16x4_f32(
                    false, af, false, bf, (short)0, acc[t], false, false);
            }
        }
    }

#pragma unroll
    for (int t = 0; t < NT; ++t) {
        const int n = n0 + t * 16 + (lane & 15);
        const float bv = bias ? bias[n] : 0.0f;
#pragma unroll
        for (int v = 0; v < 8; ++v) {
            const int m = m0 + v + ((lane >> 4) << 3);
            float val = acc[t][v] + bv;
            if (addend) val += addend[(size_t)m * ldadd + n];
            if (act == 1) val = fmaxf(val, 0.0f);
            C[(size_t)m * ldc + n] = val;
        }
    }
}

// ---------------------------------------------------------------------------
// Fused Set2Set step-2 LSTM: H2 = lstm(r1 @ WihR^T + gconst, c1_uniform).
// The 4 accumulators per wave are the 4 gates (cols j, j+160, j+320, j+480),
// so the [M,640] pre-activation tensor is never materialized.
// ---------------------------------------------------------------------------
__global__ __launch_bounds__(128)
void k_lstm2(const float* __restrict__ R,      // [M,160] = r1
             const float* __restrict__ Wih,    // [640,320]; use cols 160..319
             const float* __restrict__ gconst, // [640]
             const float* __restrict__ c1,     // [160]
             float* __restrict__ H2,           // [M,160]
             int M) {
    const int wave = threadIdx.x >> 5;
    const int lane = threadIdx.x & 31;
    const int m0 = blockIdx.x * 64 + wave * 16;
    if (m0 >= M) return;
    const int j0 = blockIdx.y * 16;
    const int mrow = lane & 15;
    const int khalf = (lane >> 4) << 1;

    v8f acc[4];
#pragma unroll
    for (int g = 0; g < 4; ++g)
#pragma unroll
        for (int v = 0; v < 8; ++v) acc[g][v] = 0.0f;

    const float* wp[4];
#pragma unroll
    for (int g = 0; g < 4; ++g)
        wp[g] = Wih + (size_t)(g * GD_ + j0 + mrow) * (2 * GD_) + GD_ + khalf;  // WihR
    const float* ap = R + (size_t)(m0 + mrow) * GD_ + khalf;

    for (int k = 0; k < GD_; k += 4) {
        v2f af = ld2g(ap); ap += 4;
#pragma unroll
        for (int g = 0; g < 4; ++g) {
            v2f bf = ld2g(wp[g]); wp[g] += 4;
            acc[g] = __builtin_amdgcn_wmma_f32_16x16x4_f32(
                false, af, false, bf, (short)0, acc[g], false, false);
        }
    }
    const int j = j0 + (lane & 15);
    const float gic = gconst[j], gfc = gconst[GD_ + j];
    const float ggc = gconst[2 * GD_ + j], goc = gconst[3 * GD_ + j];
    const float c1j = c1[j];
#pragma unroll
    for (int v = 0; v < 8; ++v) {
        const int m = m0 + v + ((lane >> 4) << 3);
        const float gi = acc[0][v] + gic;
        const float gf = acc[1][v] + gfc;
        const float gg = acc[2][v] + ggc;
        const float go = acc[3][v] + goc;
        const float c2 = sigf(gf) * c1j + sigf(gi) * tanhf(gg);
        H2[(size_t)m * GD_ + j] = sigf(go) * tanhf(c2);
    }
}

// ---------------------------------------------------------------------------
// Segment-softmax machinery (unsorted segments -> atomics).
// ---------------------------------------------------------------------------
// e[i] = dot(X[idx[i]], H[seg[i]*hstride]) (hstride=0 => uniform h vector);
// also atomicMax the segment max. One wave per item.
__global__ __launch_bounds__(256)
void k_e_dot(const float* __restrict__ X, int ld,
             const int* __restrict__ idx, const int* __restrict__ seg,
             const float* __restrict__ H, int hstride,
             float* __restrict__ e, unsigned* __restrict__ mkey, int n) {
    const int item = blockIdx.x * 8 + (threadIdx.x >> 5);
    const int lane = threadIdx.x & 31;
    if (item >= n) return;
    const int s = seg[item];
    const float* x = X + (size_t)idx[item] * ld;
    const float* h = H + (size_t)s * hstride;
    float acc = 0.0f;
    for (int d = lane; d < ld; d += 32) acc += x[d] * h[d];
#pragma unroll
    for (int off = 16; off > 0; off >>= 1) acc += __shfl_xor(acc, off, 32);
    if (lane == 0) {
        e[item] = acc;
        atomicMax(mkey + s, enc_f(acc));
    }
}

// GCN edge logits: e[i] = ndot_lrelu[src[i]]; segment max over dst.
__global__ __launch_bounds__(256)
void k_edge_max(const float* __restrict__ val, const int* __restrict__ src,
                const int* __restrict__ dst, float* __restrict__ e,
                unsigned* __restrict__ mkey, int n) {
    const int i = blockIdx.x * 256 + threadIdx.x;
    if (i >= n) return;
    const float v = val[src[i]];
    e[i] = v;
    atomicMax(mkey + dst[i], enc_f(v));
}

__global__ __launch_bounds__(256)
void k_exp(const float* __restrict__ e, const int* __restrict__ seg,
           const unsigned* __restrict__ mkey, float* __restrict__ ex,
           float* __restrict__ s, int n) {
    const int i = blockIdx.x * 256 + threadIdx.x;
    if (i >= n) return;
    const int sg = seg[i];
    const float v = expf(e[i] - dec_f(mkey[sg]));
    ex[i] = v;
    atomicAdd(s + sg, v);
}

// r[seg] += (ex/s) * X[idx]; one thread per (item, feature).
__global__ __launch_bounds__(256)
void k_raccum(const float* __restrict__ X, int ld,
              const int* __restrict__ idx, const int* __restrict__ seg,
              const float* __restrict__ ex, const float* __restrict__ s,
              float* __restrict__ r, int D, long long total) {
    const long long t = (long long)blockIdx.x * 256 + threadIdx.x;
    if (t >= total) return;
    const int i = (int)(t / D);
    const int d = (int)(t % D);
    const int sg = seg[i];
    const float a = ex[i] / (s[sg] + 1e-16f);
    atomicAdd(r + (size_t)sg * D + d, a * X[(size_t)idx[i] * ld + d]);
}

__global__ __launch_bounds__(256)
void k_count(const int* __restrict__ gidx, const int* __restrict__ aidx,
             float* __restrict__ cg, float* __restrict__ ca, int n) {
    const int i = blockIdx.x * 256 + threadIdx.x;
    if (i >= n) return;
    atomicAdd(cg + gidx[i], 1.0f);
    atomicAdd(ca + aidx[i], 1.0f);
}

__global__ __launch_bounds__(256)
void k_csum(const float* __restrict__ cond, const int* __restrict__ aidx,
            const int* __restrict__ gidx, float* __restrict__ csum, long long total) {
    const long long t = (long long)blockIdx.x * 256 + threadIdx.x;
    if (t >= total) return;
    const int i = (int)(t >> 6);
    const int d = (int)(t & 63);
    atomicAdd(csum + (size_t)gidx[i] * CD_ + d, cond[(size_t)aidx[i] * CD_ + d]);
}

__global__ __launch_bounds__(256)
void k_condg(const float* __restrict__ csum, const float* __restrict__ cnt,
             float* __restrict__ out, long long total) {
    const long long t = (long long)blockIdx.x * 256 + threadIdx.x;
    if (t >= total) return;
    out[t] = csum[t] / fmaxf(cnt[t >> 6], 1.0f);
}

// zero rows whose segment count is 0 (Set2Set present-mask)
__global__ __launch_bounds__(256)
void k_mask(float* __restrict__ buf, const float* __restrict__ cnt, int D, long long total) {
    const long long t = (long long)blockIdx.x * 256 + threadIdx.x;
    if (t >= total) return;
    if (cnt[t / D] <= 0.0f) buf[t] = 0.0f;
}

__global__ __launch_bounds__(256)
void k_film(const float* __restrict__ gamma, const float* __restrict__ xgfa,
            const float* __restrict__ beta, float* __restrict__ out, long long total) {
    const long long t = (long long)blockIdx.x * 256 + threadIdx.x;
    if (t >= total) return;
    out[t] = gamma[t] * xgfa[t] + beta[t];
}

// per-node attention logit with leaky_relu(0.2) (monotone scalar op -> can be
// applied pre-gather)
__global__ __launch_bounds__(256)
void k_ndot(const float* __restrict__ msg, const float* __restrict__ att,
            float* __restrict__ out, int n) {
    const int g = blockIdx.x * 256 + threadIdx.x;
    if (g >= n) return;
    float a = 0.0f;
    const float* m = msg + (size_t)g * G80;
    for (int d = 0; d < G80; ++d) a += m[d] * att[d];
    out[g] = (a > 0.0f) ? a : 0.2f * a;
}

// xg = concat(xgp, relu(agg + selfterm))
__global__ __launch_bounds__(256)
void k_combine(const float* __restrict__ xgp, const float* __restrict__ agg,
               const float* __restrict__ selfterm, float* __restrict__ out, long long total) {
    const long long t = (long long)blockIdx.x * 256 + threadIdx.x;
    if (t >= total) return;
    const long long g = t / GD_;
    const int d = (int)(t % GD_);
    float v;
    if (d < G80) v = xgp[g * G80 + d];
    else v = fmaxf(agg[g * G80 + (d - G80)] + selfterm[g * G80 + (d - G80)], 0.0f);
    out[t] = v;
}

// ---------------------------------------------------------------------------
extern "C" void kernel_launch(void* const* d_in, const int* in_sizes, int n_in,
                              void* d_out, int out_size, void* d_ws, size_t ws_size,
                              hipStream_t stream) {
    (void)in_sizes; (void)n_in; (void)out_size; (void)ws_size;
    const float* x_atom   = (const float*)d_in[0];
    const float* x_group  = (const float*)d_in[1];
    const float* cond     = (const float*)d_in[2];
    const float* gproj_W  = (const float*)d_in[3];
    const float* gproj_b  = (const float*)d_in[4];
    const float* fg1_W    = (const float*)d_in[5];
    const float* fg1_b    = (const float*)d_in[6];
    const float* fg2_W    = (const float*)d_in[7];
    const float* fg2_b    = (const float*)d_in[8];
    const float* fb1_W    = (const float*)d_in[9];
    const float* fb1_b    = (const float*)d_in[10];
    const float* fb2_W    = (const float*)d_in[11];
    const float* fb2_b    = (const float*)d_in[12];
    const float* apg_W    = (const float*)d_in[13];
    const float* apg_b    = (const float*)d_in[14];
    const float* s2sa_Wih = (const float*)d_in[15];
    const float* s2sa_Whh = (const float*)d_in[16];
    const float* s2sa_bih = (const float*)d_in[17];
    const float* s2sa_bhh = (const float*)d_in[18];
    const float* merge_W  = (const float*)d_in[19];
    const float* merge_b  = (const float*)d_in[20];
    const float* gmsg_W   = (const float*)d_in[21];
    const float* gmsg_b   = (const float*)d_in[22];
    const float* gself_W  = (const float*)d_in[23];
    const float* gself_b  = (const float*)d_in[24];
    const float* gcn_att  = (const float*)d_in[25];
    const float* s2sg_Wih = (const float*)d_in[26];
    const float* s2sg_Whh = (const float*)d_in[27];
    const float* s2sg_bih = (const float*)d_in[28];
    const float* s2sg_bhh = (const float*)d_in[29];
    const float* gpa_W    = (const float*)d_in[30];
    const float* gpa_b    = (const float*)d_in[31];
    const int*   atom_idx = (const int*)d_in[32];
    const int*   group_idx= (const int*)d_in[33];
    const int*   eig      = (const int*)d_in[34];
    const int*   e_src    = eig;
    const int*   e_dst    = eig + EGN;

    float* out = (float*)d_out;
    float* xa_out = out;                              // [Na,128]
    float* xg_out = out + (size_t)NA_ * AD_;          // [Gm,160]

    // ---- workspace layout (floats), bump allocator with aliasing ----------
    float* Wf = (float*)d_ws;
    size_t o = 0;
    auto alloc = [&](size_t n) { size_t r = o; o += (n + 63) & ~(size_t)63; return r; };
    const size_t o_small = alloc(1920);
    const size_t o_cntg  = alloc(GM_);
    const size_t o_cnta  = alloc(NA_);
    const size_t o_mkey  = alloc(NA_);
    const size_t o_ssum  = alloc(NA_);
    const size_t o_e     = alloc(NINC);
    const size_t o_ex    = alloc(NINC);
    const size_t o_BIG   = alloc((size_t)NA_ * GD_);   // xa_proj -> xg1 -> r_g
    const size_t o_h2a   = alloc((size_t)GM_ * GD_);   // h2_a -> gamma
    const size_t o_ra    = alloc((size_t)GM_ * GD_);   // r_a  -> beta
    const size_t o_xgfa  = alloc((size_t)GM_ * GD_);   // xgfa -> msg_all
    const size_t o_t1    = alloc((size_t)GM_ * GD_);   // t1   -> selfterm
    const size_t o_ccs   = alloc((size_t)GM_ * CD_ * 2); // csum|cond_g -> xgp
    const size_t o_agg   = alloc((size_t)GM_ * G80);
    const size_t o_ndot  = alloc(GM_);
    const size_t o_h2g   = alloc((size_t)NA_ * GD_);

    float* h1_a = Wf + o_small;          float* c1_a = h1_a + GD_;  float* gc_a = h1_a + 2 * GD_;
    float* h1_g = Wf + o_small + 960;    float* c1_g = h1_g + GD_;  float* gc_g = h1_g + 2 * GD_;
    float* cnt_g = Wf + o_cntg;
    float* cnt_a = Wf + o_cnta;
    unsigned* mkey = (unsigned*)(Wf + o_mkey);
    float* ssum = Wf + o_ssum;
    float* ebuf = Wf + o_e;
    float* exbuf = Wf + o_ex;
    float* xa_proj = Wf + o_BIG;
    float* xg1     = Wf + o_BIG;         // alias (after set2set-a done)
    float* r_g     = Wf + o_BIG;         // alias (after GCN GEMMs done)
    float* h2_a = Wf + o_h2a;  float* gamma = Wf + o_h2a;   // alias after merge
    float* r_a  = Wf + o_ra;   float* beta  = Wf + o_ra;    // alias after merge
    float* xgfa = Wf + o_xgfa; float* msg_all = Wf + o_xgfa; // alias after film
    float* t1   = Wf + o_t1;   float* selfterm = Wf + o_t1;  // alias after fb2
    float* csum = Wf + o_ccs;  float* cond_g = csum + (size_t)GM_ * CD_;
    float* xgp  = Wf + o_ccs;                               // alias after FiLM MLPs
    float* agg  = Wf + o_agg;
    float* ndot = Wf + o_ndot;
    float* h2_g = Wf + o_h2g;

    auto cdiv = [](long long a, long long b) { return (int)((a + b - 1) / b); };

    auto gemm = [&](const float* A, int lda, const float* A2, int lda2, int ksplit,
                    const float* Wt, int ldw, const float* bias,
                    const float* addend, int ldadd,
                    float* C, int ldc, int M, int N, int K, int act) {
        const int NT = (N % 64 == 0) ? 4 : ((N % 32 == 0) ? 2 : 1);
        dim3 g(cdiv(M, 64), N / (16 * NT));
        if (NT == 4)
            k_gemm<4><<<g, 128, 0, stream>>>(A, lda, A2, lda2, ksplit, Wt, ldw, bias,
                                             addend, ldadd, C, ldc, M, N, K, act);
        else if (NT == 2)
            k_gemm<2><<<g, 128, 0, stream>>>(A, lda, A2, lda2, ksplit, Wt, ldw, bias,
                                             addend, ldadd, C, ldc, M, N, K, act);
        else
            k_gemm<1><<<g, 128, 0, stream>>>(A, lda, A2, lda2, ksplit, Wt, ldw, bias,
                                             addend, ldadd, C, ldc, M, N, K, act);
    };

    // ---- 0) uniform step-1 LSTM state + gconst for both set2sets ----------
    k_small_init<<<1, 256, 0, stream>>>(s2sa_Wih, s2sa_Whh, s2sa_bih, s2sa_bhh,
                                        s2sg_Wih, s2sg_Whh, s2sg_bih, s2sg_bhh,
                                        Wf + o_small);

    // ---- 1) segment counts + cond sums ------------------------------------
    hipMemsetAsync(cnt_g, 0, sizeof(float) * GM_, stream);
    hipMemsetAsync(cnt_a, 0, sizeof(float) * NA_, stream);
    hipMemsetAsync(csum, 0, sizeof(float) * (size_t)GM_ * CD_, stream);
    k_count<<<cdiv(NINC, 256), 256, 0, stream>>>(group_idx, atom_idx, cnt_g, cnt_a, NINC);
    k_csum<<<cdiv((long long)NINC * CD_, 256), 256, 0, stream>>>(
        cond, atom_idx, group_idx, csum, (long long)NINC * CD_);

    // ---- 2) xa_proj = x_atom @ apg^T + b  [Na,160] ------------------------
    gemm(x_atom, AD_, nullptr, 0, 0, apg_W, AD_, apg_b, nullptr, 0,
         xa_proj, GD_, NA_, GD_, AD_, 0);

    // ---- set2set helper ---------------------------------------------------
    auto set2set = [&](const float* X, const int* idx, const int* seg, int B,
                       const float* h1, const float* c1v, const float* gcst,
                       const float* Wih, float* rbuf, float* h2buf, const float* cntv) {
        // step 1 (uniform h1)
        hipMemsetAsync(mkey, 0, sizeof(unsigned) * B, stream);
        hipMemsetAsync(ssum, 0, sizeof(float) * B, stream);
        k_e_dot<<<cdiv(NINC, 8), 256, 0, stream>>>(X, GD_, idx, seg, h1, 0, ebuf, mkey, NINC);
        k_exp<<<cdiv(NINC, 256), 256, 0, stream>>>(ebuf, seg, mkey, exbuf, ssum, NINC);
        hipMemsetAsync(rbuf, 0, sizeof(float) * (size_t)B * GD_, stream);
        k_raccum<<<cdiv((long long)NINC * GD_, 256), 256, 0, stream>>>(
            X, GD_, idx, seg, exbuf, ssum, rbuf, GD_, (long long)NINC * GD_);
        // step 2: h2 = lstm(r1 @ WihR^T + gconst)   (fused WMMA)
        k_lstm2<<<dim3(cdiv(B, 64), GD_ / 16), 128, 0, stream>>>(rbuf, Wih, gcst, c1v, h2buf, B);
        hipMemsetAsync(mkey, 0, sizeof(unsigned) * B, stream);
        hipMemsetAsync(ssum, 0, sizeof(float) * B, stream);
        k_e_dot<<<cdiv(NINC, 8), 256, 0, stream>>>(X, GD_, idx, seg, h2buf, GD_, ebuf, mkey, NINC);
        k_exp<<<cdiv(NINC, 256), 256, 0, stream>>>(ebuf, seg, mkey, exbuf, ssum, NINC);
        hipMemsetAsync(rbuf, 0, sizeof(float) * (size_t)B * GD_, stream);
        k_raccum<<<cdiv((long long)NINC * GD_, 256), 256, 0, stream>>>(
            X, GD_, idx, seg, exbuf, ssum, rbuf, GD_, (long long)NINC * GD_);
        // present-mask (empty segments -> exact zeros)
        k_mask<<<cdiv((long long)B * GD_, 256), 256, 0, stream>>>(h2buf, cntv, GD_, (long long)B * GD_);
        k_mask<<<cdiv((long long)B * GD_, 256), 256, 0, stream>>>(rbuf, cntv, GD_, (long long)B * GD_);
    };

    // ---- 3) set2set over groups (items: xa_proj[atom_idx], seg: group_idx)
    set2set(xa_proj, atom_idx, group_idx, GM_, h1_a, c1_a, gc_a, s2sa_Wih, r_a, h2_a, cnt_g);

    // ---- 4) merge: xgfa = [h2_a | r_a] @ merge^T + b ----------------------
    gemm(h2_a, GD_, r_a, GD_, GD_, merge_W, 2 * GD_, merge_b, nullptr, 0,
         xgfa, GD_, GM_, GD_, 2 * GD_, 0);

    // ---- 5) FiLM ----------------------------------------------------------
    k_condg<<<cdiv((long long)GM_ * CD_, 256), 256, 0, stream>>>(
        csum, cnt_g, cond_g, (long long)GM_ * CD_);
    gemm(cond_g, CD_, nullptr, 0, 0, fg1_W, CD_, fg1_b, nullptr, 0, t1, GD_, GM_, GD_, CD_, 1);
    gemm(t1, GD_, nullptr, 0, 0, fg2_W, GD_, fg2_b, nullptr, 0, gamma, GD_, GM_, GD_, GD_, 0);
    gemm(cond_g, CD_, nullptr, 0, 0, fb1_W, CD_, fb1_b, nullptr, 0, t1, GD_, GM_, GD_, CD_, 1);
    gemm(t1, GD_, nullptr, 0, 0, fb2_W, GD_, fb2_b, nullptr, 0, beta, GD_, GM_, GD_, GD_, 0);
    k_film<<<cdiv((long long)GM_ * GD_, 256), 256, 0, stream>>>(
        gamma, xgfa, beta, xg1, (long long)GM_ * GD_);

    // ---- 6) GCN (attention, additive, aggr=add) ---------------------------
    gemm(x_group, 40, nullptr, 0, 0, gproj_W, 40, gproj_b, nullptr, 0,
         xgp, G80, GM_, G80, 40, 0);
    gemm(xg1, GD_, nullptr, 0, 0, gmsg_W, GD_, gmsg_b, nullptr, 0,
         msg_all, G80, GM_, G80, GD_, 0);
    gemm(xg1, GD_, nullptr, 0, 0, gself_W, GD_, gself_b, nullptr, 0,
         selfterm, G80, GM_, G80, GD_, 0);
    k_ndot<<<cdiv(GM_, 256), 256, 0, stream>>>(msg_all, gcn_att, ndot, GM_);
    hipMemsetAsync(mkey, 0, sizeof(unsigned) * GM_, stream);
    hipMemsetAsync(ssum, 0, sizeof(float) * GM_, stream);
    k_edge_max<<<cdiv(EGN, 256), 256, 0, stream>>>(ndot, e_src, e_dst, ebuf, mkey, EGN);
    k_exp<<<cdiv(EGN, 256), 256, 0, stream>>>(ebuf, e_dst, mkey, exbuf, ssum, EGN);
    hipMemsetAsync(agg, 0, sizeof(float) * (size_t)GM_ * G80, stream);
    k_raccum<<<cdiv((long long)EGN * G80, 256), 256, 0, stream>>>(
        msg_all, G80, e_src, e_dst, exbuf, ssum, agg, G80, (long long)EGN * G80);
    k_combine<<<cdiv((long long)GM_ * GD_, 256), 256, 0, stream>>>(
        xgp, agg, selfterm, xg_out, (long long)GM_ * GD_);   // xg result -> d_out

    // ---- 7) set2set over atoms (items: xg_out[group_idx], seg: atom_idx) --
    set2set(xg_out, group_idx, atom_idx, NA_, h1_g, c1_g, gc_g, s2sg_Wih, r_g, h2_g, cnt_a);

    // ---- 8) xa_out = x_atom + [h2_g | r_g] @ gpa^T + b --------------------
    gemm(h2_g, GD_, r_g, GD_, GD_, gpa_W, 2 * GD_, gpa_b, x_atom, AD_,
         xa_out, AD_, NA_, AD_, 2 * GD_, 0);
}